// STGCNBlock_6399501271419
// MI455X (gfx1250) — compile-verified
//
#include <hip/hip_runtime.h>
#include <hip/hip_bf16.h>

// ---------------- problem constants ----------------
#define Bb   32
#define Cc   3
#define Tt   128
#define Vv   25
#define Hh   64
#define Kk   9
#define Ee   768000
#define T1   120            // Tt - Kk + 1
#define T2   112            // T1 - Kk + 1
#define Nn   96000          // Bb*T1*Vv
#define Etot (Ee + Nn)      // edges + self loops = 864000
#define OUTSZ (Bb*Cc*T2*Vv) // 268800

typedef __attribute__((ext_vector_type(2))) float v2f;
typedef __attribute__((ext_vector_type(8))) float v8f;

// order-independent (deterministic) float atomic max via int/uint trick
__device__ __forceinline__ void atomicMaxFloat(float* addr, float val) {
    if (val >= 0.f) atomicMax((int*)addr, __float_as_int(val));
    else            atomicMin((unsigned int*)addr, __float_as_uint(val));
}

// ---------------- 1) temporal conv1 + relu -> h[N,64] ----------------
__global__ void k_conv1(const float* __restrict__ x, const float* __restrict__ w1,
                        const float* __restrict__ b1, float* __restrict__ h) {
    __shared__ float sw[Hh * Cc * Kk];                 // 1728 floats
    for (int i = threadIdx.x; i < Hh * Cc * Kk; i += blockDim.x) sw[i] = w1[i];
    __syncthreads();
    int tid = blockIdx.x * blockDim.x + threadIdx.x;
    if (tid >= Nn * Hh) return;
    int f = tid & 63, n = tid >> 6;
    int v = n % Vv, t = (n / Vv) % T1, b = n / (Vv * T1);
    const float* xb = x + ((size_t)b * Cc * Tt + t) * Vv + v;
    const float* wf = sw + f * (Cc * Kk);
    float acc = b1[f];
#pragma unroll
    for (int c = 0; c < Cc; ++c)
#pragma unroll
        for (int k = 0; k < Kk; ++k)
            acc += xb[(c * Tt + k) * Vv] * wf[c * Kk + k];
    h[tid] = acc > 0.f ? acc : 0.f;
}

// ---------------- 2) xl = h@wl, xr = h@wr via V_WMMA_F32_16X16X4_F32 ------
// One wave computes one 16x16 tile of BOTH xl and xr (A fragment reused).
// A 16x4 f32 layout: lanes 0-15 rows m, VGPR0/1 = K0/K1; lanes 16-31 = K2/K3.
// B 4x16 layout mirrors C/D half-lane split: VGPR0 = rows {K0 | K2}, VGPR1 = {K1 | K3}.
// C/D 16x16 f32: VGPR r -> row (r | r+8) for lane halves, col = lane&15.
__global__ void k_gemm_wmma(const float* __restrict__ h,
                            const float* __restrict__ wl,
                            const float* __restrict__ wr,
                            float* __restrict__ xl, float* __restrict__ xr) {
    const int lane = threadIdx.x & 31;
    const int wave = threadIdx.x >> 5;   // 4 waves -> 4 N-tiles of 16
    const int m0 = blockIdx.x * 16;
    const int n0 = wave * 16;
    const int arow = m0 + (lane & 15);
    const int ak   = (lane >> 4) * 2;    // 0 or 2
    const int bcol = n0 + (lane & 15);
    const int bk   = (lane >> 4) * 2;
    v8f accl = {};
    v8f accr = {};
#pragma unroll
    for (int k0 = 0; k0 < Hh; k0 += 4) {
        const float* ap = h + (size_t)arow * Hh + k0 + ak;
        v2f a; a.x = ap[0]; a.y = ap[1];                       // contiguous float2
        v2f bl; bl.x = wl[(k0 + bk) * Hh + bcol]; bl.y = wl[(k0 + bk + 1) * Hh + bcol];
        v2f br; br.x = wr[(k0 + bk) * Hh + bcol]; br.y = wr[(k0 + bk + 1) * Hh + bcol];
        accl = __builtin_amdgcn_wmma_f32_16x16x4_f32(false, a, false, bl, (short)0, accl, false, false);
        accr = __builtin_amdgcn_wmma_f32_16x16x4_f32(false, a, false, br, (short)0, accr, false, false);
    }
    const int crow = m0 + (lane >> 4) * 8;
    const int ccol = n0 + (lane & 15);
#pragma unroll
    for (int r = 0; r < 8; ++r) {
        xl[(size_t)(crow + r) * Hh + ccol] = accl[r];
        xr[(size_t)(crow + r) * Hh + ccol] = accr[r];
    }
}

// ---------------- 3) init acc=0, z=0, m=-inf ----------------
__global__ void k_init(float* __restrict__ acc, float* __restrict__ z, float* __restrict__ m) {
    int tid = blockIdx.x * blockDim.x + threadIdx.x;
    if (tid < Nn * Hh) acc[tid] = 0.f;
    if (tid < Nn) { z[tid] = 0.f; m[tid] = -3.402823466e38f; }
}

// ---------------- 4) edge logits + segment max (8 lanes / edge) ----------
__global__ void k_edge_logit(const int* __restrict__ ei, const float* __restrict__ xl,
                             const float* __restrict__ xr, const float* __restrict__ att,
                             float* __restrict__ e_buf, float* __restrict__ m) {
    __shared__ float s_att[Hh];
    if (threadIdx.x < Hh) s_att[threadIdx.x] = att[threadIdx.x];
    __syncthreads();
    int tid = blockIdx.x * blockDim.x + threadIdx.x;
    int sub = tid & 7, e = tid >> 3;
    if (e >= Etot) return;
    int src, dst;
    if (e < Ee) { src = ei[e]; dst = ei[Ee + e]; } else { src = dst = e - Ee; }
    const float4* a4 = (const float4*)(xl + (size_t)src * Hh);
    const float4* b4 = (const float4*)(xr + (size_t)dst * Hh);
    float acc = 0.f;
#pragma unroll
    for (int q = 0; q < 2; ++q) {
        int j = sub * 2 + q;                     // float4 group 0..15
        float4 av = a4[j], bv = b4[j];
        float v0 = av.x + bv.x, v1 = av.y + bv.y, v2 = av.z + bv.z, v3 = av.w + bv.w;
        acc += (v0 > 0.f ? v0 : 0.2f * v0) * s_att[j * 4 + 0];
        acc += (v1 > 0.f ? v1 : 0.2f * v1) * s_att[j * 4 + 1];
        acc += (v2 > 0.f ? v2 : 0.2f * v2) * s_att[j * 4 + 2];
        acc += (v3 > 0.f ? v3 : 0.2f * v3) * s_att[j * 4 + 3];
    }
    acc += __shfl_xor(acc, 1, 8);
    acc += __shfl_xor(acc, 2, 8);
    acc += __shfl_xor(acc, 4, 8);
    if (sub == 0) { e_buf[e] = acc; atomicMaxFloat(&m[dst], acc); }
}

// --------- 5) exp-weighted scatter aggregation (16 lanes / edge, 4 feats) --
// acc[dst] += p * xl[src]; z[dst] += p   (alpha normalization deferred)
__global__ void k_edge_agg(const int* __restrict__ ei, const float* __restrict__ xl,
                           const float* __restrict__ e_buf, const float* __restrict__ m,
                           float* __restrict__ z, float* __restrict__ acc) {
    int tid = blockIdx.x * blockDim.x + threadIdx.x;
    int e = tid >> 4, fq = (tid & 15) * 4;
    if (e >= Etot) return;
    int src, dst;
    if (e < Ee) { src = ei[e]; dst = ei[Ee + e]; } else { src = dst = e - Ee; }
    float p = expf(e_buf[e] - m[dst]);
    if (fq == 0) atomicAdd(&z[dst], p);
    float4 a = ((const float4*)(xl + (size_t)src * Hh))[fq >> 2];
    float* o = acc + (size_t)dst * Hh + fq;
    atomicAdd(o + 0, p * a.x);
    atomicAdd(o + 1, p * a.y);
    atomicAdd(o + 2, p * a.z);
    atomicAdd(o + 3, p * a.w);
}

// ---------------- 6) node-wise normalize + bias + relu -> hn -------------
__global__ void k_node(const float* __restrict__ acc, const float* __restrict__ z,
                       const float* __restrict__ bg, float* __restrict__ hn) {
    int tid = blockIdx.x * blockDim.x + threadIdx.x;
    if (tid >= Nn * Hh) return;
    int n = tid >> 6, f = tid & 63;
    float v = acc[tid] / (z[n] + 1e-16f) + bg[f];
    hn[tid] = v > 0.f ? v : 0.f;
}

// ---------------- 7) deterministic BN stats, stage 1 ----------------
__global__ void k_stats1(const float* __restrict__ hn, float* __restrict__ partial) {
    __shared__ float ss[256], sq[256];
    int f = threadIdx.x & 63, sub = threadIdx.x >> 6;
    int base = blockIdx.x * 256;                     // 256 nodes per block
    float s = 0.f, s2 = 0.f;
    for (int i = sub; i < 256; i += 4) {
        float v = hn[(size_t)(base + i) * Hh + f];
        s += v; s2 += v * v;
    }
    ss[threadIdx.x] = s; sq[threadIdx.x] = s2;
    __syncthreads();
    if (sub == 0) {
        float ts = ss[f] + ss[64 + f] + ss[128 + f] + ss[192 + f];
        float t2 = sq[f] + sq[64 + f] + sq[128 + f] + sq[192 + f];
        partial[blockIdx.x * 128 + f]      = ts;
        partial[blockIdx.x * 128 + 64 + f] = t2;
    }
}

// ------------- 8) stats stage 2 -> scale = gamma*rsqrt(var+eps), shift ----
__global__ void k_stats2(const float* __restrict__ partial, const float* __restrict__ gamma,
                         const float* __restrict__ beta, float* __restrict__ scsh) {
    __shared__ float st[128];
    int t = threadIdx.x;
    float s = 0.f;
    for (int i = 0; i < Nn / 256; ++i) s += partial[i * 128 + t];
    st[t] = s;
    __syncthreads();
    if (t < 64) {
        float mu  = st[t] / (float)Nn;
        float var = st[64 + t] / (float)Nn - mu * mu;
        float sc  = gamma[t] * rsqrtf(var + 1e-5f);
        scsh[t]      = sc;
        scsh[64 + t] = beta[t] - mu * sc;            // shift
    }
}

// ---- 9) fold BN into conv2: w2s[(c*9+k)*64+h] = w2[c,h,k]*scale[h]; bias --
__global__ void k_w2prep(const float* __restrict__ w2, const float* __restrict__ b2,
                         const float* __restrict__ scsh, float* __restrict__ w2s,
                         float* __restrict__ b2a) {
    for (int i = threadIdx.x; i < Cc * Kk * Hh; i += blockDim.x) {
        int hf = i & 63, k = (i >> 6) % Kk, c = i / (Kk * Hh);
        w2s[i] = w2[(c * Hh + hf) * Kk + k] * scsh[hf];
    }
    if (threadIdx.x < Cc) {
        int c = threadIdx.x;
        float b = b2[c];
        for (int hf = 0; hf < Hh; ++hf) {
            float sh = scsh[64 + hf];
            for (int k = 0; k < Kk; ++k) b += sh * w2[(c * Hh + hf) * Kk + k];
        }
        b2a[c] = b;
    }
}

// ---------------- 10) temporal conv2 (BN folded) + relu -> out -----------
__global__ void k_conv2(const float* __restrict__ hn, const float* __restrict__ w2s,
                        const float* __restrict__ b2a, float* __restrict__ out) {
    __shared__ float sw[Cc * Kk * Hh];               // 1728 floats
    for (int i = threadIdx.x; i < Cc * Kk * Hh; i += blockDim.x) sw[i] = w2s[i];
    __syncthreads();
    int tid = blockIdx.x * blockDim.x + threadIdx.x;
    if (tid >= OUTSZ) return;
    int v = tid % Vv, t2 = (tid / Vv) % T2, c = (tid / (Vv * T2)) % Cc, b = tid / (Vv * T2 * Cc);
    float acc = b2a[c];
    for (int k = 0; k < Kk; ++k) {
        const float4* hv = (const float4*)(hn + (size_t)((b * T1 + t2 + k) * Vv + v) * Hh);
        const float4* wv = (const float4*)(sw + (c * Kk + k) * Hh);
#pragma unroll
        for (int q = 0; q < 16; ++q) {
            float4 a = hv[q], w = wv[q];
            acc += a.x * w.x + a.y * w.y + a.z * w.z + a.w * w.w;
        }
    }
    out[tid] = acc > 0.f ? acc : 0.f;
}

// ---------------- launcher ----------------
extern "C" void kernel_launch(void* const* d_in, const int* in_sizes, int n_in,
                              void* d_out, int out_size, void* d_ws, size_t ws_size,
                              hipStream_t stream) {
    const float* x    = (const float*)d_in[0];
    const int*   ei   = (const int*)  d_in[1];
    const float* w1   = (const float*)d_in[2];
    const float* b1   = (const float*)d_in[3];
    const float* wl   = (const float*)d_in[4];
    const float* wr   = (const float*)d_in[5];
    const float* att  = (const float*)d_in[6];
    const float* bg   = (const float*)d_in[7];
    const float* gmm  = (const float*)d_in[8];
    const float* bta  = (const float*)d_in[9];
    const float* w2   = (const float*)d_in[10];
    const float* b2   = (const float*)d_in[11];
    float* out = (float*)d_out;

    // workspace layout (floats); h reused as acc, xr reused as hn
    float* ws      = (float*)d_ws;
    float* h_acc   = ws;                            // 6,144,000
    float* xl      = ws + (size_t)Nn * Hh;          // 6,144,000
    float* xr_hn   = ws + (size_t)2 * Nn * Hh;      // 6,144,000
    float* m       = ws + (size_t)3 * Nn * Hh;      // 96,000
    float* z       = m + Nn;                        // 96,000
    float* e_buf   = z + Nn;                        // 864,000
    float* partial = e_buf + Etot;                  // 48,000
    float* scsh    = partial + (Nn / 256) * 128;    // 128
    float* w2s     = scsh + 128;                    // 1,728
    float* b2a     = w2s + Cc * Kk * Hh;            // 3

    k_conv1<<<(Nn * Hh + 255) / 256, 256, 0, stream>>>(x, w1, b1, h_acc);
    k_gemm_wmma<<<Nn / 16, 128, 0, stream>>>(h_acc, wl, wr, xl, xr_hn);
    k_init<<<(Nn * Hh + 255) / 256, 256, 0, stream>>>(h_acc, z, m);   // h dead -> acc
    k_edge_logit<<<(Etot * 8 + 255) / 256, 256, 0, stream>>>(ei, xl, xr_hn, att, e_buf, m);
    k_edge_agg<<<(Etot * 16 + 255) / 256, 256, 0, stream>>>(ei, xl, e_buf, m, z, h_acc);
    k_node<<<(Nn * Hh + 255) / 256, 256, 0, stream>>>(h_acc, z, bg, xr_hn); // xr dead -> hn
    k_stats1<<<Nn / 256, 256, 0, stream>>>(xr_hn, partial);
    k_stats2<<<1, 128, 0, stream>>>(partial, gmm, bta, scsh);
    k_w2prep<<<1, 256, 0, stream>>>(w2, b2, scsh, w2s, b2a);
    k_conv2<<<(OUTSZ + 255) / 256, 256, 0, stream>>>(xr_hn, w2s, b2a, out);
    (void)in_sizes; (void)n_in; (void)out_size; (void)ws_size;
}